// TripletLoss_22351009809014
// MI455X (gfx1250) — compile-verified
//
#include <hip/hip_runtime.h>
#include <hip/hip_bf16.h>

typedef __attribute__((ext_vector_type(2))) float v2f;
typedef __attribute__((ext_vector_type(8))) float v8f;

#define B_ROWS 8192
#define D_DIM  128
#define E_DIM  16
#define NTILES (B_ROWS / 16)
#define MARGIN_F 0.5f
#define POS_T 0.7f
#define NEG_T 0.3f

// Scratch layout (floats):
//   accum : [0, 2)
//   sq    : [256, 256+8192)
//   ewn   : [8448, 8448+131072)
//   pkW   : [139520, 139520+131072)      packed normalized expert weights
//   pkE   : [270592, 270592+1048576)     packed embeddings (WMMA fragment order)

// ---------------------------------------------------------------------------
// Kernel 1: row squared-norms of embeddings, row-normalized expert weights,
// zero the global accumulators.
// ---------------------------------------------------------------------------
__global__ __launch_bounds__(256) void prep_kernel(
    const float* __restrict__ emb, const float* __restrict__ ew,
    float* __restrict__ accum, float* __restrict__ sq, float* __restrict__ ewn)
{
    int r = blockIdx.x * 256 + threadIdx.x;
    if (r == 0) { accum[0] = 0.0f; accum[1] = 0.0f; }
    if (r >= B_ROWS) return;

    const float4* e4 = (const float4*)(emb + (size_t)r * D_DIM);
    float s = 0.0f;
#pragma unroll
    for (int i = 0; i < D_DIM / 4; ++i) {
        float4 v = e4[i];
        s += v.x*v.x + v.y*v.y + v.z*v.z + v.w*v.w;
    }
    sq[r] = s;

    const float4* w4 = (const float4*)(ew + (size_t)r * E_DIM);
    float4 w[4];
    float n = 0.0f;
#pragma unroll
    for (int i = 0; i < 4; ++i) {
        w[i] = w4[i];
        n += w[i].x*w[i].x + w[i].y*w[i].y + w[i].z*w[i].z + w[i].w*w[i].w;
    }
    float inv = 1.0f / fmaxf(sqrtf(n), 1e-8f);
    float4* o4 = (float4*)(ewn + (size_t)r * E_DIM);
#pragma unroll
    for (int i = 0; i < 4; ++i) {
        float4 v = w[i];
        v.x *= inv; v.y *= inv; v.z *= inv; v.w *= inv;
        o4[i] = v;
    }
}

// ---------------------------------------------------------------------------
// Kernel 2: repack emb / ewn into WMMA fragment order.
// For tile t, step-pair p, lane l (l = h*16 + lm):
//   float4 = row (t*16+lm), k = 8p+2h + {0,1,4,5}
//   (x,y) feeds WMMA step s=2p, (z,w) feeds step s=2p+1  (k = 4s+2h+{0,1}).
// pkE: [512 tiles][16 pairs][32 lanes] float4   (tile stride 2048 floats)
// pkW: [512 tiles][ 2 pairs][32 lanes] float4   (tile stride  256 floats)
// ---------------------------------------------------------------------------
__global__ __launch_bounds__(256) void pack_kernel(
    const float* __restrict__ emb, const float* __restrict__ ewn,
    float* __restrict__ pkE, float* __restrict__ pkW)
{
    int gid = blockIdx.x * 256 + threadIdx.x;      // 0 .. 512*16*32-1
    {
        int t  = gid >> 9;
        int p  = (gid >> 5) & 15;
        int l  = gid & 31;
        int lm = l & 15, hh = l >> 4;
        int row = t * 16 + lm;
        int k0  = 8 * p + 2 * hh;
        const float* src = emb + (size_t)row * D_DIM + k0;
        float4 v; v.x = src[0]; v.y = src[1]; v.z = src[4]; v.w = src[5];
        ((float4*)pkE)[gid] = v;
    }
    if (gid < NTILES * 2 * 32) {
        int t  = gid >> 6;
        int p  = (gid >> 5) & 1;
        int l  = gid & 31;
        int lm = l & 15, hh = l >> 4;
        int row = t * 16 + lm;
        int k0  = 8 * p + 2 * hh;
        const float* src = ewn + (size_t)row * E_DIM + k0;
        float4 v; v.x = src[0]; v.y = src[1]; v.z = src[4]; v.w = src[5];
        ((float4*)pkW)[gid] = v;
    }
}

// ---------------------------------------------------------------------------
// Kernel 3: fused tiled Gram matrices (dist + sim) via V_WMMA_F32_16X16X4_F32
// with streaming batch-hard row reduction. One block per 16-row tile,
// 8 waves split the 512 column tiles. B panels arrive as 16 coalesced
// global_load_b128 from packed fragment-order data; waves_per_eu(1) lets the
// RA keep the whole panel in flight instead of recycling one load buffer.
// ---------------------------------------------------------------------------
__global__ __launch_bounds__(256) __attribute__((amdgpu_waves_per_eu(1)))
void triplet_kernel(const float* __restrict__ pkE, const float* __restrict__ pkW,
                    const float* __restrict__ sq, float* __restrict__ accum)
{
    __shared__ float ldsPos[8][16];
    __shared__ float ldsNeg[8][16];

    const int lane = threadIdx.x & 31;
    const int wave = threadIdx.x >> 5;
    const int lm   = lane & 15;
    const int h    = lane >> 4;
    const int rowBase = blockIdx.x * 16;

    // A panel (this row tile), fragment order, kept in registers.
    const float4* aPk = (const float4*)pkE + (size_t)blockIdx.x * 512 + lane;
    float4 aE[16];
#pragma unroll
    for (int p = 0; p < 16; ++p) aE[p] = aPk[p * 32];
    const float4* aWp = (const float4*)pkW + (size_t)blockIdx.x * 64 + lane;
    float4 aW[2];
    aW[0] = aWp[0]; aW[1] = aWp[32];

    float sqI[8];
#pragma unroll
    for (int v = 0; v < 8; ++v) sqI[v] = sq[rowBase + 8 * h + v];

    float posMax[8], negMin[8];
#pragma unroll
    for (int v = 0; v < 8; ++v) { posMax[v] = -1.0f; negMin[v] = __builtin_inff(); }

    for (int ct = wave; ct < NTILES; ct += 8) {
        // ---- B panel: 16 + 2 coalesced b128 loads + 1 b32 ----
        const float4* bPk = (const float4*)pkE + (size_t)ct * 512 + lane;
        float4 bE[16];
#pragma unroll
        for (int p = 0; p < 16; ++p) bE[p] = bPk[p * 32];
        const float4* bWp = (const float4*)pkW + (size_t)ct * 64 + lane;
        float4 bW[2];
        bW[0] = bWp[0]; bW[1] = bWp[32];
        const float sqJ = sq[ct * 16 + lm];

        // ---- 36-step WMMA accumulation chain ----
        v8f accD = {};
#pragma unroll
        for (int p = 0; p < 16; ++p) {
            v2f a0 = {aE[p].x, aE[p].y}, b0 = {bE[p].x, bE[p].y};
            accD = __builtin_amdgcn_wmma_f32_16x16x4_f32(
                false, a0, false, b0, (short)0, accD, false, false);
            v2f a1 = {aE[p].z, aE[p].w}, b1 = {bE[p].z, bE[p].w};
            accD = __builtin_amdgcn_wmma_f32_16x16x4_f32(
                false, a1, false, b1, (short)0, accD, false, false);
        }
        v8f accS = {};
#pragma unroll
        for (int p = 0; p < 2; ++p) {
            v2f a0 = {aW[p].x, aW[p].y}, b0 = {bW[p].x, bW[p].y};
            accS = __builtin_amdgcn_wmma_f32_16x16x4_f32(
                false, a0, false, b0, (short)0, accS, false, false);
            v2f a1 = {aW[p].z, aW[p].w}, b1 = {bW[p].z, bW[p].w};
            accS = __builtin_amdgcn_wmma_f32_16x16x4_f32(
                false, a1, false, b1, (short)0, accS, false, false);
        }

        const int jcol = ct * 16 + lm;
#pragma unroll
        for (int v = 0; v < 8; ++v) {
            const int irow = rowBase + 8 * h + v;
            float d2   = sqI[v] + sqJ - 2.0f * accD[v];
            float dist = sqrtf(fmaxf(d2, 0.0f));
            float sim  = accS[v];
            bool nd = (irow != jcol);
            if (nd && sim > POS_T) posMax[v] = fmaxf(posMax[v], dist);
            if (nd && sim < NEG_T) negMin[v] = fminf(negMin[v], dist);
        }
    }

    // Reduce each row across its 16 lanes (xor masks < 16 stay within halves).
#pragma unroll
    for (int v = 0; v < 8; ++v) {
        float p = posMax[v], n = negMin[v];
#pragma unroll
        for (int m = 1; m < 16; m <<= 1) {
            p = fmaxf(p, __shfl_xor(p, m, 32));
            n = fminf(n, __shfl_xor(n, m, 32));
        }
        posMax[v] = p; negMin[v] = n;
    }
    if (lm == 0) {
#pragma unroll
        for (int v = 0; v < 8; ++v) {
            ldsPos[wave][8 * h + v] = posMax[v];
            ldsNeg[wave][8 * h + v] = negMin[v];
        }
    }
    __syncthreads();

    if (threadIdx.x < 16) {
        float pm = -1.0f, nm = __builtin_inff();
#pragma unroll
        for (int w = 0; w < 8; ++w) {
            pm = fmaxf(pm, ldsPos[w][threadIdx.x]);
            nm = fminf(nm, ldsNeg[w][threadIdx.x]);
        }
        bool valid = (pm >= 0.0f) && (nm < __builtin_inff());
        float rl = valid ? fmaxf(pm - nm + MARGIN_F, 0.0f) : 0.0f;
        float cv = valid ? 1.0f : 0.0f;
#pragma unroll
        for (int m = 1; m < 16; m <<= 1) {
            rl += __shfl_xor(rl, m, 32);
            cv += __shfl_xor(cv, m, 32);
        }
        if (threadIdx.x == 0) {
            atomicAdd(&accum[0], rl);
            atomicAdd(&accum[1], cv);
        }
    }
}

// ---------------------------------------------------------------------------
// Kernel 4: final scalar.
// ---------------------------------------------------------------------------
__global__ void finalize_kernel(const float* __restrict__ accum, float* __restrict__ out)
{
    float s = accum[0], c = accum[1];
    out[0] = (c > 0.0f) ? (s / c) : 0.0f;
}

extern "C" void kernel_launch(void* const* d_in, const int* in_sizes, int n_in,
                              void* d_out, int out_size, void* d_ws, size_t ws_size,
                              hipStream_t stream)
{
    const float* emb = (const float*)d_in[0];   // [8192,128] fp32
    const float* ew  = (const float*)d_in[1];   // [8192,16]  fp32
    float* ws    = (float*)d_ws;
    float* accum = ws;                          // 2 floats
    float* sq    = ws + 256;                    // 8192 floats
    float* ewn   = ws + 256 + B_ROWS;           // 131072 floats
    float* pkW   = ewn + B_ROWS * E_DIM;        // 131072 floats
    float* pkE   = pkW + B_ROWS * E_DIM;        // 1048576 floats

    prep_kernel<<<B_ROWS / 256, 256, 0, stream>>>(emb, ew, accum, sq, ewn);
    pack_kernel<<<(NTILES * 16 * 32) / 256, 256, 0, stream>>>(emb, ewn, pkE, pkW);
    triplet_kernel<<<NTILES, 256, 0, stream>>>(pkE, pkW, sq, accum);
    finalize_kernel<<<1, 1, 0, stream>>>(accum, (float*)d_out);
}